// TermRotamerScoringModule_26920855011435
// MI455X (gfx1250) — compile-verified
//
#include <hip/hip_runtime.h>

typedef float v2f __attribute__((ext_vector_type(2)));
typedef float v8f __attribute__((ext_vector_type(8)));

#define POSES 64
#define RROT  128
#define NNZ   (POSES * RROT * RROT)   // 1,048,576 rotamer pairs

// One block = 8 waves = one pose-octant. Each wave owns a 16-atom row strip
// (2 rotamers) and sweeps all 64 column tiles of its pose's 1024x1024
// atom-pair matrix using V_WMMA_F32_16X16X4_F32.
__global__ __launch_bounds__(256) void rotamer_pair_kernel(
    const float* __restrict__ coords,
    const int*   __restrict__ coord_offset_for_rot,
    float*       __restrict__ out /* scores [2, NNZ] */)
{
    __shared__ float sh[1024 * 4];           // {x, y, z, |r|^2} per atom (16 KB)

    const int tid    = threadIdx.x;
    const int lane   = tid & 31;
    const int wave   = tid >> 5;
    const int p      = blockIdx.x >> 3;                 // pose 0..63
    const int istrip = ((blockIdx.x & 7) << 3) + wave;  // row strip 0..63

    // ---- stage this pose's 1024 atoms into LDS, augmented with |r|^2 ----
    for (int a = tid; a < 1024; a += 256) {
        int rot = p * RROT + (a >> 3);
        int ga  = coord_offset_for_rot[rot] + (a & 7);   // flat atom index
        float x = coords[ga * 3 + 0];
        float y = coords[ga * 3 + 1];
        float z = coords[ga * 3 + 2];
        sh[a * 4 + 0] = x;
        sh[a * 4 + 1] = y;
        sh[a * 4 + 2] = z;
        sh[a * 4 + 3] = x * x + y * y + z * z;
    }
    __syncthreads();

    const bool lo  = lane < 16;
    const int  n16 = lane & 15;

    // ---- A operand, fixed per wave. Row m = istrip*16 + n16.
    // 32-bit A 16x4 layout: lanes 0-15 hold K=0,1 ; lanes 16-31 hold K=2,3.
    // A[m][*] = { -2x, -2y, -2z, |r_m|^2 }
    const int m_atom = istrip * 16 + n16;
    v2f av = *reinterpret_cast<const v2f*>(&sh[m_atom * 4 + (lo ? 0 : 2)]);
    v2f A;
    A.x = -2.0f * av.x;                    // K0: -2x  | K2: -2z
    A.y = lo ? (-2.0f * av.y) : av.y;      // K1: -2y  | K3: |r|^2

    const float kLog2e = 1.44269504088896340736f;

    #pragma unroll 2
    for (int jt = 0; jt < 64; ++jt) {
        // ---- B operand: column n = jt*16 + n16, B[*][n] = { x, y, z, 1 }
        const int n_atom = jt * 16 + n16;
        v2f bv = *reinterpret_cast<const v2f*>(&sh[n_atom * 4 + (lo ? 0 : 2)]);
        float sqw = sh[n_atom * 4 + 3];    // |r_n|^2, lane-invariant across VGPRs
        v2f B;
        B.x = bv.x;                        // K0: x | K2: z
        B.y = lo ? bv.y : 1.0f;            // K1: y | K3: 1

        // D = |r_m|^2 - 2 r_m . r_n
        v8f acc = {};
        acc = __builtin_amdgcn_wmma_f32_16x16x4_f32(
            /*neg_a=*/false, A, /*neg_b=*/false, B,
            /*c_mod=*/(short)0, acc, /*reuse_a=*/false, /*reuse_b=*/false);

        // ---- postprocess: d2 -> 1/d2 and exp(-sqrt(d2)), in-lane M reduction
        float s0 = 0.0f, s1 = 0.0f;
        #pragma unroll
        for (int v = 0; v < 8; ++v) {
            float d2 = fmaxf(acc[v] + sqw, 1e-2f);
            s0 += __builtin_amdgcn_rcpf(d2);
            float t = __builtin_amdgcn_sqrtf(d2);
            s1 += __builtin_amdgcn_exp2f(t * -kLog2e);
        }

        // ---- reduce N axis: sum over 8 lanes of each column sub-block
        #pragma unroll
        for (int m = 1; m <= 4; m <<= 1) {
            s0 += __shfl_xor(s0, m, 32);
            s1 += __shfl_xor(s1, m, 32);
        }

        // lanes 0,8,16,24 hold the 4 rotamer-pair sums of this tile
        if ((lane & 7) == 0) {
            int ri  = istrip * 2 + (lane >> 4);
            int rj  = jt * 2 + ((lane >> 3) & 1);
            int idx = p * (RROT * RROT) + ri * RROT + rj;
            out[idx]       = s0;   // LJ-like term
            out[NNZ + idx] = s1;   // LK-like term
        }
    }
}

// indices [3, NNZ]: (pose, global rot_i, global rot_j) as int32
__global__ __launch_bounds__(256) void indices_kernel(
    const int* __restrict__ pose_for_rot,
    const int* __restrict__ rot_offset_for_pose,
    int*       __restrict__ idx_out)
{
    int i = blockIdx.x * blockDim.x + threadIdx.x;
    if (i >= NNZ) return;
    int pr   = i >> 7;                 // i / R  -> rotamer-row index in [0, P*R)
    int p    = pose_for_rot[pr];
    int ri   = pr & (RROT - 1);
    int rj   = i & (RROT - 1);
    int base = rot_offset_for_pose[p];
    idx_out[i]           = p;
    idx_out[NNZ + i]     = base + ri;
    idx_out[2 * NNZ + i] = base + rj;
}

extern "C" void kernel_launch(void* const* d_in, const int* in_sizes, int n_in,
                              void* d_out, int out_size, void* d_ws, size_t ws_size,
                              hipStream_t stream) {
    const float* coords               = (const float*)d_in[0];
    const int*   coord_offset_for_rot = (const int*)d_in[1];
    const int*   pose_for_rot         = (const int*)d_in[2];
    const int*   rot_offset_for_pose  = (const int*)d_in[3];

    float* out = (float*)d_out;

    // 64 poses x 8 blocks, 8 waves each -> 4096 waves, 256 pair outputs/wave
    rotamer_pair_kernel<<<512, 256, 0, stream>>>(coords, coord_offset_for_rot, out);

    int* idx_out = (int*)(out + 2 * (size_t)NNZ);
    indices_kernel<<<(NNZ + 255) / 256, 256, 0, stream>>>(
        pose_for_rot, rot_offset_for_pose, idx_out);
}